// GNNDebugger_16338055594576
// MI455X (gfx1250) — compile-verified
//
#include <hip/hip_runtime.h>
#include <hip/hip_bf16.h>

// ---------------------------------------------------------------------------
// GCN 2-layer forward for MI455X (gfx1250, wave32).
//   h1   = x @ W1                      (WMMA f32 16x16x4, async-LDS staged A)
//   a1   = relu( Dhat (A+I) Dhat h1 + b1 )   (atomic f32 scatter, L2-resident)
//   h2   = a1 @ W2
//   out  = Dhat (A+I) Dhat h2 + b2
// ---------------------------------------------------------------------------

typedef __attribute__((ext_vector_type(2))) float v2f;
typedef __attribute__((ext_vector_type(8))) float v8f;
typedef int v4i_vs __attribute__((vector_size(16)));  // matches builtin param type

#define IN_C 128
#define LDS_STRIDE 132  // 128 + 4 pad: conflict-free strided column reads

#if __has_builtin(__builtin_amdgcn_global_load_async_to_lds_b128)
#define HAVE_ASYNC_LDS 1
#else
#define HAVE_ASYNC_LDS 0
#endif

// Builtin signature (from clang diagnostic): param0 = v4i addrspace(1)* (global,
// printed as "__device__"), param1 = LDS side, then imm offset, imm cpol.
// Generic -> as(1)/as(3) via integer cast; low 32 bits of a generic LDS
// address are the LDS offset on AMDGPU.
#define TO_GLB_V4(p) ((__attribute__((address_space(1))) v4i_vs*)(unsigned long long)(p))
#define TO_LDS_V4(p) ((__attribute__((address_space(3))) v4i_vs*)(unsigned int)(unsigned long long)(p))

static __device__ __forceinline__ void wait_asynccnt0() {
#if __has_builtin(__builtin_amdgcn_s_wait_asynccnt)
  __builtin_amdgcn_s_wait_asynccnt(0);
#else
  asm volatile("s_wait_asynccnt 0" ::: "memory");
#endif
}

// ---------------- degree / normalization ----------------

__global__ void k_fill_deg(float* __restrict__ deg, int n) {
  int i = blockIdx.x * blockDim.x + threadIdx.x;
  if (i < n) deg[i] = 1.0f;  // self-loop
}

__global__ void k_count_deg(const long long* __restrict__ dst,
                            float* __restrict__ deg, int nEdges) {
  int e = blockIdx.x * blockDim.x + threadIdx.x;
  if (e < nEdges) {
    long long d = dst[e];
    __hip_atomic_fetch_add(&deg[d], 1.0f, __ATOMIC_RELAXED,
                           __HIP_MEMORY_SCOPE_AGENT);
  }
}

__global__ void k_to_dinv(float* __restrict__ deg, int n) {
  int i = blockIdx.x * blockDim.x + threadIdx.x;
  if (i < n) deg[i] = rsqrtf(deg[i]);
}

// ---------------- WMMA GEMM:  H[N,OUT] = A[N,128] * W[128,OUT] ----------------
// NT = OUT/16 column tiles; block = NT waves. Each wave owns one 16-col panel
// of W in 64 VGPRs. Blocks grid-stride over 16-row A tiles, which stream
// through double-buffered LDS via global_load_async_to_lds_b128 (ASYNCcnt).

template <int NT>
__global__ void __launch_bounds__(NT * 32)
k_gcn_gemm(const float* __restrict__ A, const float* __restrict__ W,
           float* __restrict__ H, int nRows) {
  constexpr int OUT     = NT * 16;
  constexpr int THREADS = NT * 32;
  constexpr int CHUNKS  = (16 * IN_C) / 4;      // float4 chunks per tile
  constexpr int ITERS   = CHUNKS / THREADS;     // chunks per thread

#if HAVE_ASYNC_LDS
  __shared__ float lds[2][16 * LDS_STRIDE];
#else
  __shared__ float lds[1][16 * LDS_STRIDE];
#endif

  const int t    = threadIdx.x;
  const int wave = t >> 5;
  const int lane = t & 31;
  const int half = lane >> 4;   // 0: lanes 0-15, 1: lanes 16-31
  const int m    = lane & 15;
  const int n0   = wave * 16;

  // Preload B panel: per K-step kk, rows 4kk+2*half (+1), col n0+m.
  v2f bfrag[32];
#pragma unroll
  for (int kk = 0; kk < 32; ++kk) {
    const int kr = 4 * kk + 2 * half;
    v2f b;
    b.x = W[kr * OUT + n0 + m];
    b.y = W[(kr + 1) * OUT + n0 + m];
    bfrag[kk] = b;
  }

  const int ntiles = nRows >> 4;  // N = 100000 = 6250 * 16, exact

#if HAVE_ASYNC_LDS
  // ---- double-buffered async pipeline ----
  int tile = blockIdx.x;
  if (tile < ntiles) {
    const int row0 = tile << 4;
#pragma unroll
    for (int i = 0; i < ITERS; ++i) {
      const int q = t + i * THREADS;
      const int r = q >> 5;
      const int c = (q & 31) << 2;
      __builtin_amdgcn_global_load_async_to_lds_b128(
          TO_GLB_V4(A + (size_t)(row0 + r) * IN_C + c),
          TO_LDS_V4(&lds[0][r * LDS_STRIDE + c]), 0, 0);
    }
  }

  int cur = 0;
  for (; tile < ntiles; tile += gridDim.x) {
    wait_asynccnt0();   // this wave's stage of lds[cur] landed
    __syncthreads();    // everyone's stage landed; prior reads of lds[1-cur] done

    const int nxt = tile + gridDim.x;
    if (nxt < ntiles) {
      const int nrow0 = nxt << 4;
#pragma unroll
      for (int i = 0; i < ITERS; ++i) {
        const int q = t + i * THREADS;
        const int r = q >> 5;
        const int c = (q & 31) << 2;
        __builtin_amdgcn_global_load_async_to_lds_b128(
            TO_GLB_V4(A + (size_t)(nrow0 + r) * IN_C + c),
            TO_LDS_V4(&lds[1 - cur][r * LDS_STRIDE + c]), 0, 0);
      }
    }

    v8f acc = {};
#pragma unroll
    for (int kk = 0; kk < 32; ++kk) {
      const int kb = 4 * kk + 2 * half;
      const v2f a  = *(const v2f*)&lds[cur][m * LDS_STRIDE + kb];
      acc = __builtin_amdgcn_wmma_f32_16x16x4_f32(
          false, a, false, bfrag[kk], (short)0, acc, false, false);
    }

    const int row0 = tile << 4;
#pragma unroll
    for (int r = 0; r < 8; ++r) {
      H[(size_t)(row0 + r + 8 * half) * OUT + n0 + m] = acc[r];
    }
    cur ^= 1;
  }
#else
  // ---- synchronous fallback, vectorized staging ----
  for (int tile = blockIdx.x; tile < ntiles; tile += gridDim.x) {
    const int row0 = tile << 4;
    __syncthreads();
#pragma unroll
    for (int i = 0; i < ITERS; ++i) {
      const int q = t + i * THREADS;
      const int r = q >> 5;
      const int c = (q & 31) << 2;
      *(float4*)&lds[0][r * LDS_STRIDE + c] =
          *(const float4*)(A + (size_t)(row0 + r) * IN_C + c);
    }
    __syncthreads();

    v8f acc = {};
#pragma unroll
    for (int kk = 0; kk < 32; ++kk) {
      const int kb = 4 * kk + 2 * half;
      const v2f a  = *(const v2f*)&lds[0][m * LDS_STRIDE + kb];
      acc = __builtin_amdgcn_wmma_f32_16x16x4_f32(
          false, a, false, bfrag[kk], (short)0, acc, false, false);
    }
#pragma unroll
    for (int r = 0; r < 8; ++r) {
      H[(size_t)(row0 + r + 8 * half) * OUT + n0 + m] = acc[r];
    }
  }
#endif
}

// ---------------- aggregation ----------------

// out[i][c] = bias[c] + h[i][c] * dinv[i]^2   (self-loop term + bias), float4
template <int C>
__global__ void k_init_out(float4* __restrict__ out, const float4* __restrict__ h,
                           const float4* __restrict__ bias,
                           const float* __restrict__ dinv, int n) {
  constexpr int C4 = C / 4;
  size_t i = (size_t)blockIdx.x * blockDim.x + threadIdx.x;
  if (i >= (size_t)n * C4) return;
  const int node = (int)(i / C4);
  const int c4   = (int)(i % C4);
  const float di = dinv[node];
  const float s  = di * di;
  const float4 hv = h[i];
  const float4 bv = bias[c4];
  out[i] = make_float4(bv.x + hv.x * s, bv.y + hv.y * s,
                       bv.z + hv.z * s, bv.w + hv.w * s);
}

// One wave per edge: out[dst] += h[src] * dinv[src]*dinv[dst]
template <int C>
__global__ void k_edge_agg(const long long* __restrict__ src,
                           const long long* __restrict__ dst,
                           const float* __restrict__ dinv,
                           const float* __restrict__ h,
                           float* __restrict__ out, int nEdges) {
  const int eid = blockIdx.x * (blockDim.x >> 5) + (threadIdx.x >> 5);
  if (eid >= nEdges) return;
  const int lane = threadIdx.x & 31;

  const long long s = src[eid];
  const long long d = dst[eid];
  const float norm  = dinv[s] * dinv[d];

  if constexpr (C == 128) {
    const float4 hv = *(const float4*)(h + (size_t)s * C + lane * 4);
    float* op       = out + (size_t)d * C + lane * 4;
    __hip_atomic_fetch_add(&op[0], hv.x * norm, __ATOMIC_RELAXED, __HIP_MEMORY_SCOPE_AGENT);
    __hip_atomic_fetch_add(&op[1], hv.y * norm, __ATOMIC_RELAXED, __HIP_MEMORY_SCOPE_AGENT);
    __hip_atomic_fetch_add(&op[2], hv.z * norm, __ATOMIC_RELAXED, __HIP_MEMORY_SCOPE_AGENT);
    __hip_atomic_fetch_add(&op[3], hv.w * norm, __ATOMIC_RELAXED, __HIP_MEMORY_SCOPE_AGENT);
  } else {
    const float2 hv = *(const float2*)(h + (size_t)s * C + lane * 2);
    float* op       = out + (size_t)d * C + lane * 2;
    __hip_atomic_fetch_add(&op[0], hv.x * norm, __ATOMIC_RELAXED, __HIP_MEMORY_SCOPE_AGENT);
    __hip_atomic_fetch_add(&op[1], hv.y * norm, __ATOMIC_RELAXED, __HIP_MEMORY_SCOPE_AGENT);
  }
}

__global__ void k_relu(float4* __restrict__ a, size_t n4) {
  size_t i = (size_t)blockIdx.x * blockDim.x + threadIdx.x;
  if (i < n4) {
    float4 v = a[i];
    a[i] = make_float4(fmaxf(v.x, 0.f), fmaxf(v.y, 0.f),
                       fmaxf(v.z, 0.f), fmaxf(v.w, 0.f));
  }
}

// ---------------- launch ----------------

extern "C" void kernel_launch(void* const* d_in, const int* in_sizes, int n_in,
                              void* d_out, int out_size, void* d_ws,
                              size_t ws_size, hipStream_t stream) {
  const float* x        = (const float*)d_in[0];
  const long long* ei   = (const long long*)d_in[1];
  const float* W1       = (const float*)d_in[2];
  const float* b1       = (const float*)d_in[3];
  const float* W2       = (const float*)d_in[4];
  const float* b2       = (const float*)d_in[5];
  float* out            = (float*)d_out;

  const int N = in_sizes[0] / IN_C;     // 100000
  const int E = in_sizes[1] / 2;        // 3200000
  const long long* src = ei;            // edge_index[0]
  const long long* dst = ei + E;        // edge_index[1]

  // Workspace layout: dinv [N] | bufA [N*128] | bufB [N*128]
  float* wsf        = (float*)d_ws;
  const size_t nPad = (((size_t)N + 511) / 512) * 512;
  float* dinv       = wsf;
  float* bufA       = wsf + nPad;
  float* bufB       = bufA + (size_t)N * 128;

  const int T = 256;

  // Degrees (with self-loop) -> dinv = rsqrt(deg)
  k_fill_deg<<<(N + T - 1) / T, T, 0, stream>>>(dinv, N);
  k_count_deg<<<(E + T - 1) / T, T, 0, stream>>>(dst, dinv, E);
  k_to_dinv<<<(N + T - 1) / T, T, 0, stream>>>(dinv, N);

  // Layer 1
  k_gcn_gemm<8><<<2048, 256, 0, stream>>>(x, W1, bufA, N);              // h1
  {
    const int total4 = N * (128 / 4);
    k_init_out<128><<<(total4 + T - 1) / T, T, 0, stream>>>(
        (float4*)bufB, (const float4*)bufA, (const float4*)b1, dinv, N);
  }
  k_edge_agg<128><<<(E + 7) / 8, 256, 0, stream>>>(src, dst, dinv, bufA, bufB, E);
  {
    const size_t n4 = (size_t)N * 128 / 4;
    k_relu<<<(int)((n4 + T - 1) / T), T, 0, stream>>>((float4*)bufB, n4);
  }

  // Layer 2
  k_gcn_gemm<4><<<2048, 128, 0, stream>>>(bufB, W2, bufA, N);           // h2
  {
    const int total4 = N * (64 / 4);
    k_init_out<64><<<(total4 + T - 1) / T, T, 0, stream>>>(
        (float4*)out, (const float4*)bufA, (const float4*)b2, dinv, N);
  }
  k_edge_agg<64><<<(E + 7) / 8, 256, 0, stream>>>(src, dst, dinv, bufA, out, E);
}